// CRF_1597727834175
// MI455X (gfx1250) — compile-verified
//
#include <hip/hip_runtime.h>
#include <hip/hip_bf16.h>
#include <stdint.h>

#define NUM_LABELS 128
#define BATCH 256
#define SEQ 1024
#define NEG -10000.0f

// ---- CDNA5 async global->LDS support (probe-confirmed present) ----
#if defined(__has_builtin)
#  if __has_builtin(__builtin_amdgcn_global_load_async_to_lds_b128)
#    define HAVE_ASYNC_B128 1
#  endif
#  if __has_builtin(__builtin_amdgcn_s_wait_asynccnt)
#    define HAVE_WAIT_ASYNC 1
#  endif
#endif

typedef int v4i __attribute__((ext_vector_type(4)));
typedef __attribute__((address_space(1))) v4i GV4;  // global int4
typedef __attribute__((address_space(3))) v4i LV4;  // LDS int4

// Copy 16 bytes global -> LDS per lane (async on CDNA5).
__device__ __forceinline__ void async_copy16(const void* g, void* l) {
#if HAVE_ASYNC_B128
  __builtin_amdgcn_global_load_async_to_lds_b128((GV4*)g, (LV4*)l, 0, 0);
#else
  *reinterpret_cast<uint4*>(l) = *reinterpret_cast<const uint4*>(g);
#endif
}

// Copy 512 bytes (one emission row of 128 floats) global -> LDS.
// Call with lanes 0..31 of one wave: each lane moves 16 bytes.
__device__ __forceinline__ void async_row512(const float* g, float* l, int lane) {
  async_copy16(g + lane * 4, l + lane * 4);
}

__device__ __forceinline__ void wait_async0() {
#if HAVE_WAIT_ASYNC
  __builtin_amdgcn_s_wait_asynccnt(0);
#else
  __asm__ volatile("s_wait_asynccnt 0" ::: "memory");
#endif
}

__device__ __forceinline__ void wait_async3() {
#if HAVE_WAIT_ASYNC
  __builtin_amdgcn_s_wait_asynccnt(3);
#else
  __asm__ volatile("s_wait_asynccnt 3" ::: "memory");
#endif
}

// ---- Kernel 1: passthrough copy of inputs -> out (bandwidth bound) ----
__global__ void crf_copy_kernel(const float4* __restrict__ src,
                                float4* __restrict__ dst, int n4) {
  int idx = blockIdx.x * blockDim.x + threadIdx.x;
  int stride = gridDim.x * blockDim.x;
  for (int i = idx; i < n4; i += stride) dst[i] = src[i];
}

// ---- Kernel 2: fused Viterbi forward + final argmax + backtrack ----
// Grid: BATCH blocks of NUM_LABELS threads (4 waves, wave32).
__global__ __launch_bounds__(NUM_LABELS) void crf_viterbi_kernel(
    const float* __restrict__ inputs,   // [B, T, L]
    const float* __restrict__ trans,    // [L, L]
    float* __restrict__ out_scores,     // [B]
    float* __restrict__ out_paths,      // [B, T] (stored as float)
    uint8_t* __restrict__ roads) {      // [B, T, L] workspace backpointers
  const int b = blockIdx.x;
  const int j = threadIdx.x;  // current tag handled by this thread

  __shared__ __align__(16) float v_lds[2][NUM_LABELS];     // double-buffered
  __shared__ __align__(16) float emit_lds[4][NUM_LABELS];  // 4-deep async ring
  __shared__ __align__(16) uint8_t road_lds[64][NUM_LABELS];  // 8 KB chunk
  __shared__ int path_lds[64];
  __shared__ int cur_ptr;

  // Transition column j lives in registers (128 VGPRs/lane, fully unrolled).
  float col[NUM_LABELS];
#pragma unroll
  for (int i = 0; i < NUM_LABELS; ++i) col[i] = trans[i * NUM_LABELS + j];

  // v0: NEG everywhere except tags 0 and 3.
  v_lds[0][j] = (j == 0 || j == 3) ? 0.0f : NEG;

  const float* emit_base = inputs + (size_t)b * SEQ * NUM_LABELS;
  uint8_t* road_base = roads + (size_t)b * SEQ * NUM_LABELS;

  // Prime the async emission pipeline 3 deep (wave 0 only).
  if (j < 32) {
    async_row512(emit_base + 0 * NUM_LABELS, &emit_lds[0][0], j);
    async_row512(emit_base + 1 * NUM_LABELS, &emit_lds[1][0], j);
    async_row512(emit_base + 2 * NUM_LABELS, &emit_lds[2][0], j);
  }
  __syncthreads();

  for (int t = 0; t < SEQ; ++t) {
    const int rb = t & 1;  // v read buffer
    if (j < 32) {
      if (t + 3 < SEQ) {
        async_row512(emit_base + (size_t)(t + 3) * NUM_LABELS,
                     &emit_lds[(t + 3) & 3][0], j);
        wait_async3();  // <=3 outstanding => emit[t] has landed
      } else {
        wait_async0();  // tail: drain everything
      }
    }
    __syncthreads();  // emit[t] visible to all waves; v[t] published

    // max-plus reduction over previous tags i with 4 independent
    // accumulator chains (high ILP); first-occurrence argmax semantics.
    float b0 = -3.4e38f, b1 = -3.4e38f, b2 = -3.4e38f, b3 = -3.4e38f;
    int x0 = 0, x1 = 1, x2 = 2, x3 = 3;
#pragma unroll
    for (int i = 0; i < NUM_LABELS; i += 4) {
      float4 vv = *reinterpret_cast<const float4*>(&v_lds[rb][i]);  // broadcast
      float s0 = vv.x + col[i + 0]; if (s0 > b0) { b0 = s0; x0 = i + 0; }
      float s1 = vv.y + col[i + 1]; if (s1 > b1) { b1 = s1; x1 = i + 1; }
      float s2 = vv.z + col[i + 2]; if (s2 > b2) { b2 = s2; x2 = i + 2; }
      float s3 = vv.w + col[i + 3]; if (s3 > b3) { b3 = s3; x3 = i + 3; }
    }
    // Merge chains; on equal values prefer the smaller index (replicates
    // jnp.argmax first-occurrence tie-breaking exactly).
    float best = b0; int bidx = x0;
    if (b1 > best || (b1 == best && x1 < bidx)) { best = b1; bidx = x1; }
    if (b2 > best || (b2 == best && x2 < bidx)) { best = b2; bidx = x2; }
    if (b3 > best || (b3 == best && x3 < bidx)) { best = b3; bidx = x3; }

    const float vnew = best + emit_lds[t & 3][j];
    road_base[(size_t)t * NUM_LABELS + j] = (uint8_t)bidx;
    v_lds[rb ^ 1][j] = vnew;  // write other buffer; one barrier per step
  }
  __syncthreads();

  // Final score + last-state argmax (first occurrence), thread 0 serial scan.
  const int fb = SEQ & 1;  // buffer holding v_T
  if (j == 0) {
    float bs = v_lds[fb][0];
    int bl = 0;
#pragma unroll 4
    for (int i = 1; i < NUM_LABELS; ++i) {
      float v = v_lds[fb][i];
      if (v > bs) { bs = v; bl = i; }
    }
    out_scores[b] = bs;
    cur_ptr = bl;
  }

  // Our own road stores must be visible to this block's loads.
  __threadfence();
  __syncthreads();

  // Backtrack in 64-step chunks: async coalesced loads into LDS, then a short
  // dependent walk in LDS (avoids a 1024-deep dependent global-load chain).
  const int CH = 64;
  for (int cs = SEQ - CH; cs >= 0; cs -= CH) {
    const uint4* src =
        reinterpret_cast<const uint4*>(road_base + (size_t)cs * NUM_LABELS);
    uint4* dst = reinterpret_cast<uint4*>(&road_lds[0][0]);
#pragma unroll
    for (int k = 0; k < 4; ++k)  // 128 threads x 4 x 16B = 8192 B, coalesced
      async_copy16(src + j + k * NUM_LABELS, dst + j + k * NUM_LABELS);
    wait_async0();    // per-wave: its async chunk loads have landed
    __syncthreads();  // whole 8 KB chunk visible to all waves
    if (j == 0) {
      int p = cur_ptr;
      for (int tt = CH - 1; tt >= 0; --tt) {
        p = road_lds[tt][p];   // path[t] = road[t][path[t+1]]
        path_lds[tt] = p;
      }
      cur_ptr = p;
    }
    __syncthreads();
    if (j < CH)
      out_paths[(size_t)b * SEQ + cs + j] = (float)path_lds[j];
    __syncthreads();  // protect road_lds/path_lds before next chunk
  }
}

extern "C" void kernel_launch(void* const* d_in, const int* in_sizes, int n_in,
                              void* d_out, int out_size, void* d_ws,
                              size_t ws_size, hipStream_t stream) {
  (void)in_sizes; (void)n_in; (void)out_size; (void)ws_size;
  const float* inputs = (const float*)d_in[0];   // [256,1024,128] f32
  const float* trans = (const float*)d_in[1];    // [128,128] f32

  float* out = (float*)d_out;
  float* out_inputs = out;                                      // 33,554,432
  float* out_scores = out + (size_t)BATCH * SEQ * NUM_LABELS;   // 256
  float* out_paths = out_scores + BATCH;                        // 262,144

  uint8_t* roads = (uint8_t*)d_ws;  // needs B*T*L = 33,554,432 bytes

  const int n4 = BATCH * SEQ * NUM_LABELS / 4;
  crf_copy_kernel<<<2048, 256, 0, stream>>>(
      (const float4*)inputs, (float4*)out_inputs, n4);

  crf_viterbi_kernel<<<BATCH, NUM_LABELS, 0, stream>>>(
      inputs, trans, out_scores, out_paths, roads);
}